// cmdnet_kb_36936718745661
// MI455X (gfx1250) — compile-verified
//
#include <hip/hip_runtime.h>
#include <math.h>

// ---------------- problem constants (match reference) ----------------
constexpr int Bsz = 4096;
constexpr int Ndim = 128;
constexpr int Kdim = 64;
constexpr int ITER = 64;
constexpr int Mdim = 4;

// LDS strides (floats), padded for bank-conflict avoidance (64 banks)
constexpr int HT_LD = 80;   // +2 rows => +160 floats => +32 banks (disjoint half-lanes)
constexpr int HH_LD = 65;   // row k, elem l => bank (k+l)%64 : conflict-free row scan

typedef __attribute__((ext_vector_type(2))) float v2f;
typedef __attribute__((ext_vector_type(8))) float v8f;
typedef __attribute__((ext_vector_type(4))) int   v4i;

#if defined(__has_builtin)
#if __has_builtin(__builtin_amdgcn_global_load_async_to_lds_b128) && \
    __has_builtin(__builtin_amdgcn_s_wait_asynccnt)
#define HAVE_ASYNC_LDS 1
#endif
#endif

__global__ __launch_bounds__(64) void cmdnet_fused_kernel(
    const float* __restrict__ yt,      // [B,N]
    const float* __restrict__ Ht,      // [B,N,K]
    const float* __restrict__ sigmat0, // [B]
    const float* __restrict__ taui,    // [IT]
    const float* __restrict__ delta,   // [IT]
    float* __restrict__ ft_out,        // [B,K,M]
    float* __restrict__ xt_out)        // [B,K]
{
    // sHt (phase 1) and sHH (phase 2+) are aliased: sHt is dead once the Gram
    // accumulators and yH are in registers. ~42 KB LDS -> 7 blocks per WGP.
    __shared__ float sBuf[Ndim * HT_LD];          // 40 KB
    __shared__ float syt[Ndim];
    __shared__ float syH[Kdim];
    __shared__ float sxt[2][Kdim];                // double-buffered: 1 barrier/iter
    __shared__ float staui[ITER];
    __shared__ float sdelta[ITER];

    float (*sHt)[HT_LD] = reinterpret_cast<float (*)[HT_LD]>(sBuf);
    float (*sHH)[HH_LD] = reinterpret_cast<float (*)[HH_LD]>(sBuf);  // 64*65 < 128*80

    const int b    = blockIdx.x;
    const int tid  = threadIdx.x;       // 0..63 (2 waves)
    const int wave = tid >> 5;          // 0..1
    const int lane = tid & 31;
    const int kh   = lane >> 4;         // half-wave
    const int mm   = lane & 15;

    const float* __restrict__ Hb = Ht + (size_t)b * Ndim * Kdim;
    const float* __restrict__ yb = yt + (size_t)b * Ndim;

    // ---- Phase 0: stage Ht[b] (128x64 f32) into LDS ----
#if HAVE_ASYNC_LDS
    for (int i = tid; i < (Ndim * Kdim) / 4; i += 64) {
        const int row = i >> 4;
        const int c4  = i & 15;
        __builtin_amdgcn_global_load_async_to_lds_b128(
            (__attribute__((address_space(1))) v4i*)(Hb + (size_t)i * 4),
            (__attribute__((address_space(3))) v4i*)(&sHt[row][c4 * 4]),
            0, 0);
    }
#else
    for (int i = tid; i < (Ndim * Kdim) / 4; i += 64) {
        const int row = i >> 4;
        const int c4  = i & 15;
        float4 v = reinterpret_cast<const float4*>(Hb)[i];
        *reinterpret_cast<float4*>(&sHt[row][c4 * 4]) = v;
    }
#endif
    for (int i = tid; i < Ndim; i += 64) syt[i] = yb[i];
    staui[tid]  = taui[tid];
    sdelta[tid] = delta[tid];
#if HAVE_ASYNC_LDS
    __builtin_amdgcn_s_wait_asynccnt(0);
#endif
    __syncthreads();

    // ---- Phase 1: HH = Ht^T Ht via V_WMMA_F32_16X16X4_F32 ----
    // 64x64 output = 4x4 tiles of 16x16; wave w owns tile rows {2w, 2w+1}.
    // A-frag (16x4 of Ht^T) and B-frag (4x16 of Ht) share one lane layout:
    //   frag.x = Ht[n0 + 2*kh][col + mm], frag.y = Ht[n0 + 2*kh + 1][col + mm].
    // A is loaded at a runtime *address* (16*(2w+a)) to avoid register selects.
    v8f acc[2][4];
#pragma unroll
    for (int a = 0; a < 2; ++a)
#pragma unroll
        for (int li = 0; li < 4; ++li) acc[a][li] = (v8f)0.0f;

    const int acol0 = 16 * (2 * wave)     + mm;
    const int acol1 = 16 * (2 * wave + 1) + mm;

#pragma unroll 4
    for (int nc = 0; nc < Ndim / 4; ++nc) {
        const int n0 = nc * 4 + 2 * kh;
        const float* r0 = &sHt[n0][0];
        const float* r1 = &sHt[n0 + 1][0];
        v2f bf[4], af[2];
#pragma unroll
        for (int j = 0; j < 4; ++j) {
            bf[j].x = r0[16 * j + mm];
            bf[j].y = r1[16 * j + mm];
        }
        af[0].x = r0[acol0];  af[0].y = r1[acol0];
        af[1].x = r0[acol1];  af[1].y = r1[acol1];
#pragma unroll
        for (int a = 0; a < 2; ++a) {
#pragma unroll
            for (int li = 0; li < 4; ++li) {
                acc[a][li] = __builtin_amdgcn_wmma_f32_16x16x4_f32(
                    false, af[a], false, bf[li], (short)0, acc[a][li], false, false);
            }
        }
    }

    // ---- Phase 1b: yH[k] = sum_n Ht[n][k] * yt[n] (thread k owns column k) ----
    float accy = 0.0f;
#pragma unroll 8
    for (int n = 0; n < Ndim; ++n) accy = fmaf(sHt[n][tid], syt[n], accy);

    __syncthreads();   // all sHt reads done before sHH (aliased) is written

    // store accumulators -> sHH ; C/D layout: lanes 0-15 rows vr, 16-31 rows vr+8
#pragma unroll
    for (int a = 0; a < 2; ++a) {
        const int krow = 16 * (2 * wave + a) + 8 * kh;
#pragma unroll
        for (int li = 0; li < 4; ++li) {
#pragma unroll
            for (int vr = 0; vr < 8; ++vr) {
                sHH[krow + vr][16 * li + mm] = acc[a][li][vr];
            }
        }
    }
    syH[tid] = accy;
    __syncthreads();

    // ---- Phase 2: 64 CMDNet iterations; thread t owns k = t, G[4] in regs ----
    const float MC0 = -3.0f / 2.2360679774997896964f;  // -3/sqrt(5)
    const float MC1 = -1.0f / 2.2360679774997896964f;
    const float MC2 =  1.0f / 2.2360679774997896964f;
    const float MC3 =  3.0f / 2.2360679774997896964f;
    const float LOGA = -1.3862943611198906188f;        // log(0.25), uniform over M

    const float sg   = sigmat0[b];
    const float sig2 = sg * sg;
    const float yHk  = syH[tid];

    float G0 = 0.0f, G1 = 0.0f, G2 = 0.0f, G3 = 0.0f;

    for (int it = 0; it < ITER; ++it) {
        const float s  = fabsf(staui[it]);
        const float sf = (it == 0) ? 1.0f : s;
        const float di = sdelta[it];

        float p0 = (LOGA + G0) * sf, p1 = (LOGA + G1) * sf;
        float p2 = (LOGA + G2) * sf, p3 = (LOGA + G3) * sf;
        float mx = fmaxf(fmaxf(p0, p1), fmaxf(p2, p3));
        float e0 = __expf(p0 - mx), e1 = __expf(p1 - mx);
        float e2 = __expf(p2 - mx), e3 = __expf(p3 - mx);
        float inv = 1.0f / (e0 + e1 + e2 + e3);
        float f0 = e0 * inv, f1 = e1 * inv, f2 = e2 * inv, f3 = e3 * inv;

        float xk = f0 * MC0 + f1 * MC1 + f2 * MC2 + f3 * MC3;

        // double-buffered broadcast vector: iteration it uses buffer it&1;
        // the next iteration writes the other buffer, so one barrier suffices.
        const int pb = it & 1;
        sxt[pb][tid] = xk;
        __syncthreads();

        // xHH[k] = sum_l xt[l] * HH[l][k] (HH symmetric -> scan row k; sxt broadcast)
        float xhh = 0.0f;
#pragma unroll 8
        for (int l = 0; l < Kdim; ++l) xhh = fmaf(sxt[pb][l], sHH[tid][l], xhh);

        const float r = xhh - yHk;

        float g;
        g = sig2 * (1.0f - __expf(-G0)) + s * f0 * (MC0 - xk) * r;  G0 -= di * g;
        g = sig2 * (1.0f - __expf(-G1)) + s * f1 * (MC1 - xk) * r;  G1 -= di * g;
        g = sig2 * (1.0f - __expf(-G2)) + s * f2 * (MC2 - xk) * r;  G2 -= di * g;
        g = sig2 * (1.0f - __expf(-G3)) + s * f3 * (MC3 - xk) * r;  G3 -= di * g;
    }

    // ---- Phase 3: final soft output with s_last ----
    {
        const float s = fabsf(staui[ITER - 1]);
        float p0 = (LOGA + G0) * s, p1 = (LOGA + G1) * s;
        float p2 = (LOGA + G2) * s, p3 = (LOGA + G3) * s;
        float mx = fmaxf(fmaxf(p0, p1), fmaxf(p2, p3));
        float e0 = __expf(p0 - mx), e1 = __expf(p1 - mx);
        float e2 = __expf(p2 - mx), e3 = __expf(p3 - mx);
        float inv = 1.0f / (e0 + e1 + e2 + e3);
        float f0 = e0 * inv, f1 = e1 * inv, f2 = e2 * inv, f3 = e3 * inv;
        float xk = f0 * MC0 + f1 * MC1 + f2 * MC2 + f3 * MC3;

        const size_t kidx = (size_t)b * Kdim + tid;
        reinterpret_cast<float4*>(ft_out)[kidx] = make_float4(f0, f1, f2, f3);
        xt_out[kidx] = xk;
    }
}

extern "C" void kernel_launch(void* const* d_in, const int* in_sizes, int n_in,
                              void* d_out, int out_size, void* d_ws, size_t ws_size,
                              hipStream_t stream) {
    const float* yt    = (const float*)d_in[0];
    const float* Ht    = (const float*)d_in[1];
    const float* sig   = (const float*)d_in[2];
    const float* taui  = (const float*)d_in[3];
    const float* delta = (const float*)d_in[4];

    float* ft_out = (float*)d_out;                        // [B,K,M]
    float* xt_out = ft_out + (size_t)Bsz * Kdim * Mdim;   // [B,K]

    hipLaunchKernelGGL(cmdnet_fused_kernel, dim3(Bsz), dim3(64), 0, stream,
                       yt, Ht, sig, taui, delta, ft_out, xt_out);
}